// GCNDiffPool_14740327760547
// MI455X (gfx1250) — compile-verified
//
#include <hip/hip_runtime.h>
#include <hip/hip_bf16.h>

typedef __attribute__((ext_vector_type(2))) float v2f;
typedef __attribute__((ext_vector_type(8))) float v8f;

#define NN      9600
#define NPG     150
#define NPGP    160     // padded to multiple of 16
#define HID     64
#define CL      25
#define CLP     32      // clusters padded
#define BG      64
#define NCLS    10

// ---------------------------------------------------------------------------
// Pad Wa [64x25] -> Wap [64x32], ba [25] -> bap [32]
// ---------------------------------------------------------------------------
__global__ void k_padWa(const float* __restrict__ Wa, const float* __restrict__ ba,
                        float* __restrict__ Wap, float* __restrict__ bap) {
    int t = blockIdx.x * blockDim.x + threadIdx.x;
    if (t < HID * CLP) {
        int r = t / CLP, c = t % CLP;
        Wap[t] = (c < CL) ? Wa[r * CL + c] : 0.0f;
    }
    if (t < CLP) bap[t] = (t < CL) ? ba[t] : 0.0f;
}

// ---------------------------------------------------------------------------
// Node degrees: d_isqrt[i] = rsqrt(max(1 + sum_j A_g[i,j], 1e-12))
// ---------------------------------------------------------------------------
__global__ void k_degree(const float* __restrict__ a, float* __restrict__ dis) {
    int i = blockIdx.x * blockDim.x + threadIdx.x;
    if (i >= NN) return;
    int g = i / NPG;
    const float* row = a + (size_t)i * NN + (size_t)g * NPG;
    float s = 1.0f;  // self loop
    for (int j = 0; j < NPG; ++j) s += row[j];
    dis[i] = rsqrtf(fmaxf(s, 1e-12f));
}

// ---------------------------------------------------------------------------
// Generic dense GEMM: C[M, NT*16] strip per wave.
//   C = rowscale ⊙ (A @ B) (+bias)(+relu).  fp32 WMMA 16x16x4.
// A fragment loaded once per k-step, reused across NT B tiles.
// ---------------------------------------------------------------------------
template <int NT>
__global__ __launch_bounds__(32) void k_gemm(
    const float* __restrict__ A, const float* __restrict__ B, float* __restrict__ C,
    int K, int lda, int ldb, int ldc,
    const float* __restrict__ rowscale, const float* __restrict__ bias, int relu) {
    int lane = threadIdx.x & 31;
    int hf = lane >> 4, lr = lane & 15;
    int m0 = blockIdx.x * 16, n0 = blockIdx.y * 16 * NT;
    v8f acc[NT];
#pragma unroll
    for (int t = 0; t < NT; ++t) acc[t] = (v8f){};
    const float* ap = A + (size_t)(m0 + lr) * lda;
    for (int k0 = 0; k0 < K; k0 += 4) {
        int ka = k0 + 2 * hf;  // even
        v2f av = *(const v2f*)(ap + ka);
        const float* b0 = B + (size_t)ka * ldb + n0 + lr;
#pragma unroll
        for (int t = 0; t < NT; ++t) {
            v2f bv;
            bv.x = b0[t * 16];
            bv.y = b0[ldb + t * 16];
            acc[t] = __builtin_amdgcn_wmma_f32_16x16x4_f32(false, av, false, bv,
                                                           (short)0, acc[t], false, false);
        }
    }
#pragma unroll
    for (int t = 0; t < NT; ++t) {
        int col = n0 + t * 16 + lr;
#pragma unroll
        for (int r = 0; r < 8; ++r) {
            int row = m0 + 8 * hf + r;
            float v = acc[t][r];
            if (bias) v += bias[col];
            if (rowscale) v *= rowscale[row];
            if (relu) v = fmaxf(v, 0.0f);
            C[(size_t)row * ldc + col] = v;
        }
    }
}

// ---------------------------------------------------------------------------
// Per-graph block aggregation on the 150x150 diagonal blocks of `a`:
//   normalize=1: Y[i] = relu?( dis[i]*((A_g@X)[i] + X[i]) + bias )   (GCN step)
//   normalize=0: Y[i] = (A_g@X)[i]                                   (plain A@S)
// Each wave: 16 x (NT*16) strip; the A-block row fragment is read ONCE
// per k-step and reused across all NT column tiles.
// ---------------------------------------------------------------------------
template <int NT>
__global__ __launch_bounds__(32) void k_blockagg(
    const float* __restrict__ a, const float* __restrict__ X, float* __restrict__ Y,
    const float* __restrict__ dis, const float* __restrict__ bias,
    int ldx, int relu, int normalize) {
    int lane = threadIdx.x & 31;
    int hf = lane >> 4, lr = lane & 15;
    int g = blockIdx.x, m0 = blockIdx.y * 16;
    int rb = g * NPG;
    int mrow = m0 + lr;
    bool mok = mrow < NPG;
    const float* arow = a + (size_t)(rb + mrow) * NN + rb;
    v8f acc[NT];
#pragma unroll
    for (int t = 0; t < NT; ++t) acc[t] = (v8f){};
    for (int k0 = 0; k0 < NPGP; k0 += 4) {
        int ka = k0 + 2 * hf;          // even; ka<150 implies ka+1<150
        bool kok = ka < NPG;
        v2f av;
        av.x = (mok && kok) ? arow[ka] : 0.0f;
        av.y = (mok && kok) ? arow[ka + 1] : 0.0f;
        const float* xk = X + (size_t)(rb + ka) * ldx + lr;
#pragma unroll
        for (int t = 0; t < NT; ++t) {
            v2f bv;
            bv.x = kok ? xk[t * 16] : 0.0f;
            bv.y = kok ? xk[ldx + t * 16] : 0.0f;
            acc[t] = __builtin_amdgcn_wmma_f32_16x16x4_f32(false, av, false, bv,
                                                           (short)0, acc[t], false, false);
        }
    }
#pragma unroll
    for (int t = 0; t < NT; ++t) {
        int col = t * 16 + lr;
#pragma unroll
        for (int r = 0; r < 8; ++r) {
            int row = m0 + 8 * hf + r;
            if (row < NPG) {
                int i = rb + row;
                float v = acc[t][r];
                if (normalize) { v += X[(size_t)i * ldx + col]; v *= dis[i]; }
                if (bias) v += bias[col];
                if (relu) v = fmaxf(v, 0.0f);
                Y[(size_t)i * ldx + col] = v;
            }
        }
    }
}

// ---------------------------------------------------------------------------
// Per-graph transposed pooling: Y_g[m, n] = sum_k S_g[k, m] * B_g[k, n]
// (S stored [NN, 32]).  Used for Zp = S^T Z and Ap = S^T (A S).
// ---------------------------------------------------------------------------
template <int NT>
__global__ __launch_bounds__(32) void k_poolT(
    const float* __restrict__ S, const float* __restrict__ Bm, float* __restrict__ Y,
    int ldb, int ldy, int rowstride, int mmax) {
    int lane = threadIdx.x & 31;
    int hf = lane >> 4, lr = lane & 15;
    int g = blockIdx.x, m0 = blockIdx.y * 16;
    int rb = g * NPG;
    v8f acc[NT];
#pragma unroll
    for (int t = 0; t < NT; ++t) acc[t] = (v8f){};
    for (int k0 = 0; k0 < NPGP; k0 += 4) {
        int ka = k0 + 2 * hf;  // even
        bool kok = ka < NPG;
        v2f av;
        av.x = kok ? S[(size_t)(rb + ka) * CLP + m0 + lr] : 0.0f;
        av.y = kok ? S[(size_t)(rb + ka + 1) * CLP + m0 + lr] : 0.0f;
        const float* bk = Bm + (size_t)(rb + ka) * ldb + lr;
#pragma unroll
        for (int t = 0; t < NT; ++t) {
            v2f bv;
            bv.x = kok ? bk[t * 16] : 0.0f;
            bv.y = kok ? bk[ldb + t * 16] : 0.0f;
            acc[t] = __builtin_amdgcn_wmma_f32_16x16x4_f32(false, av, false, bv,
                                                           (short)0, acc[t], false, false);
        }
    }
#pragma unroll
    for (int t = 0; t < NT; ++t) {
        int col = t * 16 + lr;
#pragma unroll
        for (int r = 0; r < 8; ++r) {
            int m = m0 + 8 * hf + r;
            if (m < mmax) Y[((size_t)g * rowstride + m) * ldy + col] = acc[t][r];
        }
    }
}

// ---------------------------------------------------------------------------
// Row-wise softmax over first 25 of 32 cols, in place; zero padded cols.
// ---------------------------------------------------------------------------
__global__ void k_softmax(float* __restrict__ S) {
    int i = blockIdx.x * blockDim.x + threadIdx.x;
    if (i >= NN) return;
    float* row = S + (size_t)i * CLP;
    float m = -3.4e38f;
    float v[CL];
    for (int c = 0; c < CL; ++c) { v[c] = row[c]; m = fmaxf(m, v[c]); }
    float s = 0.0f;
    for (int c = 0; c < CL; ++c) { v[c] = __expf(v[c] - m); s += v[c]; }
    float inv = 1.0f / s;
    for (int c = 0; c < CL; ++c) row[c] = v[c] * inv;
    for (int c = CL; c < CLP; ++c) row[c] = 0.0f;
}

// ---------------------------------------------------------------------------
// Pooled-graph degrees from Ap [BG][32][32] (rows/cols >=25 are exact zeros)
// ---------------------------------------------------------------------------
__global__ void k_pooldeg(const float* __restrict__ Ap, float* __restrict__ disp) {
    int t = blockIdx.x * blockDim.x + threadIdx.x;
    if (t >= BG * CL) return;
    int g = t / CL, i = t % CL;
    const float* row = Ap + (size_t)g * CLP * CLP + (size_t)i * CLP;
    float s = 1.0f;
    for (int j = 0; j < CL; ++j) s += row[j];
    disp[t] = rsqrtf(fmaxf(s, 1e-12f));
}

// ---------------------------------------------------------------------------
// Pooled GCN aggregation: Hp[i] = relu(disp[i]*((Ap_g@Zs)[i] + Zs[i]) + bp)
// Ap [BG][32][32]; Zs/Hp compact [BG*25, 64].  16 x 64 strip per wave.
// ---------------------------------------------------------------------------
__global__ __launch_bounds__(32) void k_poolagg(
    const float* __restrict__ Ap, const float* __restrict__ Zs, float* __restrict__ Hp,
    const float* __restrict__ disp, const float* __restrict__ bp) {
    const int NT = 4;
    int lane = threadIdx.x & 31;
    int hf = lane >> 4, lr = lane & 15;
    int g = blockIdx.x, m0 = blockIdx.y * 16;
    const float* ag = Ap + (size_t)g * CLP * CLP;
    v8f acc[NT];
#pragma unroll
    for (int t = 0; t < NT; ++t) acc[t] = (v8f){};
    for (int k0 = 0; k0 < CLP; k0 += 4) {
        int ka = k0 + 2 * hf;  // even; note CL=25 is odd -> separate masks
        v2f av = *(const v2f*)(ag + (size_t)(m0 + lr) * CLP + ka);
        const float* zk = Zs + ((size_t)g * CL + ka) * HID + lr;
#pragma unroll
        for (int t = 0; t < NT; ++t) {
            v2f bv;
            bv.x = (ka < CL) ? zk[t * 16] : 0.0f;
            bv.y = (ka + 1 < CL) ? zk[HID + t * 16] : 0.0f;
            acc[t] = __builtin_amdgcn_wmma_f32_16x16x4_f32(false, av, false, bv,
                                                           (short)0, acc[t], false, false);
        }
    }
#pragma unroll
    for (int t = 0; t < NT; ++t) {
        int col = t * 16 + lr;
#pragma unroll
        for (int r = 0; r < 8; ++r) {
            int m = m0 + 8 * hf + r;
            if (m < CL) {
                int i = g * CL + m;
                float v = acc[t][r] + Zs[(size_t)i * HID + col];
                v = disp[i] * v + bp[col];
                Hp[(size_t)i * HID + col] = fmaxf(v, 0.0f);
            }
        }
    }
}

// ---------------------------------------------------------------------------
// Readout: G[g,h] = sum_c Hp[g*25+c, h]; logits = G @ Wc + bc
// ---------------------------------------------------------------------------
__global__ __launch_bounds__(64) void k_readout(
    const float* __restrict__ Hp, const float* __restrict__ Wc,
    const float* __restrict__ bc, float* __restrict__ out) {
    __shared__ float sh[HID];
    int g = blockIdx.x, t = threadIdx.x;
    float s = 0.0f;
    for (int c = 0; c < CL; ++c) s += Hp[((size_t)g * CL + c) * HID + t];
    sh[t] = s;
    __syncthreads();
    if (t < NCLS) {
        float l = bc[t];
        for (int h = 0; h < HID; ++h) l += sh[h] * Wc[h * NCLS + t];
        out[g * NCLS + t] = l;
    }
}

// ---------------------------------------------------------------------------
extern "C" void kernel_launch(void* const* d_in, const int* in_sizes, int n_in,
                              void* d_out, int out_size, void* d_ws, size_t ws_size,
                              hipStream_t stream) {
    const float* x  = (const float*)d_in[0];
    const float* a  = (const float*)d_in[1];
    const float* W1 = (const float*)d_in[4];
    const float* b1 = (const float*)d_in[5];
    const float* W2 = (const float*)d_in[6];
    const float* b2 = (const float*)d_in[7];
    const float* Wa = (const float*)d_in[8];
    const float* ba = (const float*)d_in[9];
    const float* Wp = (const float*)d_in[10];
    const float* bp = (const float*)d_in[11];
    const float* Wc = (const float*)d_in[12];
    const float* bc = (const float*)d_in[13];
    float* out = (float*)d_out;

    float* w = (float*)d_ws;
    float* dis  = w;                 w += NN;              // 9600
    float* disp = w;                 w += BG * CL;         // 1600
    float* Wap  = w;                 w += HID * CLP;       // 2048
    float* bap  = w;                 w += CLP;             // 32
    float* P    = w;                 w += NN * HID;        // scratch A
    float* Q    = w;                 w += NN * HID;        // scratch B
    float* R    = w;                 w += NN * HID;        // Z2
    float* Zp   = w;                 w += BG * CL * HID;   // [1600,64]
    float* Ap   = w;                 w += BG * CLP * CLP;  // [64,32,32]
    float* Zs   = w;                 w += BG * CL * HID;
    float* Hp   = w;                 w += BG * CL * HID;

    k_padWa<<<(HID * CLP + 255) / 256, 256, 0, stream>>>(Wa, ba, Wap, bap);
    k_degree<<<(NN + 255) / 256, 256, 0, stream>>>(a, dis);

    // Layer 1: Xs = dis ⊙ (x @ W1);  Z1 = relu(dis⊙((A@Xs)+Xs) + b1)
    k_gemm<4><<<dim3(NN / 16, 1), 32, 0, stream>>>(x, W1, P, 128, 128, HID, HID, dis, nullptr, 0);
    k_blockagg<4><<<dim3(BG, NPGP / 16), 32, 0, stream>>>(a, P, Q, dis, b1, HID, 1, 1);

    // Layer 2 -> Z2 in R
    k_gemm<4><<<dim3(NN / 16, 1), 32, 0, stream>>>(Q, W2, P, HID, HID, HID, HID, dis, nullptr, 0);
    k_blockagg<4><<<dim3(BG, NPGP / 16), 32, 0, stream>>>(a, P, R, dis, b2, HID, 1, 1);

    // Assignment: SA = GCN(a, Z2, Wap, bap) (cols padded to 32), softmax -> S in Q
    k_gemm<2><<<dim3(NN / 16, 1), 32, 0, stream>>>(R, Wap, P, HID, HID, CLP, CLP, dis, nullptr, 0);
    k_blockagg<2><<<dim3(BG, NPGP / 16), 32, 0, stream>>>(a, P, Q, dis, bap, CLP, 0, 1);
    k_softmax<<<(NN + 255) / 256, 256, 0, stream>>>(Q);

    // Zp = S^T @ Z2   (per graph, [25,64])
    k_poolT<4><<<dim3(BG, CLP / 16), 32, 0, stream>>>(Q, R, Zp, HID, HID, CL, CL);
    // AS = A @ S (per-graph blocks) -> P ; Ap = S^T @ AS  [64,32,32]
    k_blockagg<2><<<dim3(BG, NPGP / 16), 32, 0, stream>>>(a, Q, P, nullptr, nullptr, CLP, 0, 0);
    k_poolT<2><<<dim3(BG, CLP / 16), 32, 0, stream>>>(Q, P, Ap, CLP, CLP, CLP, CLP);

    // Pooled GCN
    k_pooldeg<<<(BG * CL + 255) / 256, 256, 0, stream>>>(Ap, disp);
    k_gemm<4><<<dim3(BG * CL / 16, 1), 32, 0, stream>>>(Zp, Wp, Zs, HID, HID, HID, HID, disp, nullptr, 0);
    k_poolagg<<<dim3(BG, CLP / 16), 32, 0, stream>>>(Ap, Zs, Hp, disp, bp);

    // Readout + classifier
    k_readout<<<BG, HID, 0, stream>>>(Hp, Wc, bc, out);
}